// ClusterVarLoss_72705206387243
// MI455X (gfx1250) — compile-verified
//
#include <hip/hip_runtime.h>
#include <hip/hip_bf16.h>
#include <stdint.h>

// ---------------------------------------------------------------------------
// ClusterVarLoss for MI455X (gfx1250).
// embedding: [N, 256] f32, cluster_out: [N, 256] f32 -> scalar f32.
// Memory-bound (512 MB @ 23.3 TB/s ~ 22 us). Strategy:
//   P1: wave-per-row argmax (coalesced b128 NT loads, wave32 shuffle reduce)
//   P2: per-WG 256KB LDS segment-sum table (fits CDNA5's 320KB WGP LDS),
//       fed by a 3-deep global_load_async_to_lds_b128 pipeline; per-block
//       partials written as plain streaming stores (no global atomics).
//   P3: grid reduction of partials.
//   P4: single-WG finalize; per-cluster ||mean||^2 via fp32 WMMA Gram tiles
//       (v_wmma_f32_16x16x4_f32 diagonals), then weighted variance dot.
// ---------------------------------------------------------------------------

typedef float v2f __attribute__((ext_vector_type(2)));
typedef float v4f __attribute__((ext_vector_type(4)));
typedef float v8f __attribute__((ext_vector_type(8)));

#define DDIM 256              // embedding dim == num clusters
#define ACC_F   65536         // LDS floats: accumulator [256][256]
#define CNT_F   65536         // LDS float offset: counts[256]
#define SM_F    65792         // LDS float offset: second-moment sums[256]
#define STAGE_F 66048         // LDS float offset: staging buffers
#define TILE_ROWS 16
#define TILE_FLOATS (TILE_ROWS * DDIM)      // 4096 floats
#define TILE_BYTES  (TILE_FLOATS * 4)       // 16384 bytes
#define NBUF 3
#define LDS_BYTES ((STAGE_F + NBUF * TILE_FLOATS) * 4)  // 313344 B <= 320KB

__device__ __forceinline__ void lds_fadd(float* p, float v) {
    __hip_atomic_fetch_add(p, v, __ATOMIC_RELAXED, __HIP_MEMORY_SCOPE_WORKGROUP);
}

// Async copy: 16B per lane, GVS mode (SGPR base + 32-bit VGPR byte offset).
__device__ __forceinline__ void async_b128(uint32_t lds_byte_addr,
                                           uint32_t gbyte_off,
                                           const float* gbase) {
    asm volatile("global_load_async_to_lds_b128 %0, %1, %2"
                 :: "v"(lds_byte_addr), "v"(gbyte_off), "s"(gbase)
                 : "memory");
}

// ---------------------------------------------------------------------------
// Pass 1: argmax over each cluster_out row. One wave32 per row.
// ---------------------------------------------------------------------------
__global__ void __launch_bounds__(256)
k_argmax(const float* __restrict__ co, int* __restrict__ labels, int N) {
    const int gwave = (int)((blockIdx.x * blockDim.x + threadIdx.x) >> 5);
    const int lane  = threadIdx.x & 31;
    if (gwave >= N) return;

    const v4f* p = (const v4f*)(co + (size_t)gwave * DDIM);
    v4f x0 = __builtin_nontemporal_load(&p[lane]);       // cols 4*lane..+3
    v4f x1 = __builtin_nontemporal_load(&p[lane + 32]);  // cols 128+4*lane..+3

    float best = x0[0];
    int   bi   = lane * 4;
    #pragma unroll
    for (int j = 1; j < 4; ++j)
        if (x0[j] > best) { best = x0[j]; bi = lane * 4 + j; }
    #pragma unroll
    for (int j = 0; j < 4; ++j)
        if (x1[j] > best) { best = x1[j]; bi = 128 + lane * 4 + j; }

    // wave reduce: max value, tie -> lowest index (matches jnp.argmax)
    #pragma unroll
    for (int m = 16; m > 0; m >>= 1) {
        float ov = __shfl_xor(best, m, 32);
        int   oi = __shfl_xor(bi,   m, 32);
        if (ov > best || (ov == best && oi < bi)) { best = ov; bi = oi; }
    }
    if (lane == 0) labels[gwave] = bi;
}

// ---------------------------------------------------------------------------
// Pass 2: per-workgroup LDS segment accumulation with async staging pipeline.
// 1024 threads = 32 waves; 2 waves per sample row (128 cols each).
// ---------------------------------------------------------------------------
__global__ void __launch_bounds__(1024)
k_accum(const float* __restrict__ emb, const int* __restrict__ labels,
        float* __restrict__ psum, float* __restrict__ pcnt,
        float* __restrict__ psm, int ntiles, int tiles_per_blk, int Ntot) {
    extern __shared__ float lds[];
    const int tid = threadIdx.x;

    for (int i = tid; i < STAGE_F; i += 1024) lds[i] = 0.0f;
    __syncthreads();

    const uint32_t lds_base   = (uint32_t)(uintptr_t)lds;   // low 32b = LDS offset
    const uint32_t stage_base = lds_base + STAGE_F * 4u;
    const uint32_t glimit     = (uint32_t)Ntot * (DDIM * 4u);

    const int t0 = blockIdx.x * tiles_per_blk;
    int t1 = t0 + tiles_per_blk; if (t1 > ntiles) t1 = ntiles;
    const int nt = t1 - t0;

    // prologue: fill up to 3 staging buffers
    for (int k = 0; k < NBUF && k < nt; ++k) {
        const uint32_t goff  = (uint32_t)(t0 + k) * TILE_BYTES + (uint32_t)tid * 16u;
        const uint32_t laddr = stage_base + (uint32_t)k * TILE_BYTES + (uint32_t)tid * 16u;
        if (goff < glimit) async_b128(laddr, goff, emb);
    }

    const int wv   = tid >> 5;
    const int lane = tid & 31;
    const int row  = wv >> 1;   // 0..15
    const int half = wv & 1;    // 0/1 -> which 128 columns

    for (int ti = 0; ti < nt; ++ti) {
        const int t   = t0 + ti;
        const int buf = ti % NBUF;

        if (nt - ti > NBUF) asm volatile("s_wait_asynccnt 2" ::: "memory");
        else                asm volatile("s_wait_asynccnt 0" ::: "memory");
        __syncthreads();    // tile t fully staged by all waves

        const int n = t * TILE_ROWS + row;
        if (n < Ntot) {     // uniform per wave
            const int lab = labels[n];
            const float* srow = &lds[STAGE_F + buf * TILE_FLOATS + row * DDIM + half * 128];
            float* arow = &lds[lab * DDIM + half * 128];
            float ss = 0.0f;
            #pragma unroll
            for (int j = 0; j < 4; ++j) {
                const float v = srow[j * 32 + lane];   // conflict-free, stride-1
                ss = fmaf(v, v, ss);
                lds_fadd(&arow[j * 32 + lane], v);     // ds_add_f32
            }
            #pragma unroll
            for (int m = 16; m > 0; m >>= 1) ss += __shfl_xor(ss, m, 32);
            if (lane == 0) {
                lds_fadd(&lds[SM_F + lab], ss);
                if (half == 0) lds_fadd(&lds[CNT_F + lab], 1.0f);
            }
        }
        __syncthreads();    // all waves done consuming buf before refill

        if (ti + NBUF < nt) {
            const int tn = t0 + ti + NBUF;
            const uint32_t goff  = (uint32_t)tn * TILE_BYTES + (uint32_t)tid * 16u;
            const uint32_t laddr = stage_base + (uint32_t)buf * TILE_BYTES + (uint32_t)tid * 16u;
            if (goff < glimit) async_b128(laddr, goff, emb);
        }
    }
    __syncthreads();

    // stream per-block partials (plain stores; no global atomics)
    const int b = blockIdx.x;
    for (int i = tid; i < ACC_F; i += 1024)
        psum[(size_t)b * ACC_F + i] = lds[i];
    if (tid < DDIM) {
        pcnt[b * DDIM + tid] = lds[CNT_F + tid];
        psm [b * DDIM + tid] = lds[SM_F  + tid];
    }
}

// ---------------------------------------------------------------------------
// Pass 3: reduce per-block partials -> cluster_sum[65536], counts[256], sm[256]
// ---------------------------------------------------------------------------
__global__ void __launch_bounds__(256)
k_reduce(const float* __restrict__ psum, const float* __restrict__ pcnt,
         const float* __restrict__ psm, float* __restrict__ csum,
         float* __restrict__ cnt, float* __restrict__ sm, int nblk) {
    const int i = blockIdx.x * blockDim.x + threadIdx.x;
    if (i < ACC_F) {
        float s = 0.0f;
        for (int b = 0; b < nblk; ++b) s += psum[(size_t)b * ACC_F + i];
        csum[i] = s;
    } else if (i < ACC_F + DDIM) {
        const int k = i - ACC_F;
        float s = 0.0f;
        for (int b = 0; b < nblk; ++b) s += pcnt[b * DDIM + k];
        cnt[k] = s;
    } else if (i < ACC_F + 2 * DDIM) {
        const int k = i - ACC_F - DDIM;
        float s = 0.0f;
        for (int b = 0; b < nblk; ++b) s += psm[b * DDIM + k];
        sm[k] = s;
    }
}

// ---------------------------------------------------------------------------
// Pass 4: finalize. 16 waves; wave w computes Gram tile of clusters
// [16w, 16w+16) via v_wmma_f32_16x16x4_f32 (A == B == mean tile, symmetric).
// Diagonal of C = ||mean_k||^2 in exact fp32. Then weighted variance dot.
// ---------------------------------------------------------------------------
__global__ void __launch_bounds__(512)
k_final(const float* __restrict__ csum, const float* __restrict__ cnt,
        const float* __restrict__ sm, float* __restrict__ out, int Ntot) {
    __shared__ float gram[16][8][32];
    __shared__ float normsq[DDIM];
    __shared__ float red[512];

    const int tid  = threadIdx.x;
    const int w    = tid >> 5;
    const int lane = tid & 31;

    // A-matrix f32 16x4 layout: lanes 0-15 M=lane (K=0 in v0, K=1 in v1);
    // lanes 16-31 M=lane-16 (K=2 in v0, K=3 in v1). B 4x16 mirrors it, and
    // for a symmetric Gram (B = A^T over the same tile) b == a per lane.
    const int c   = w * 16 + (lane & 15);
    const int klo = (lane >> 4) * 2;            // 0 or 2
    const float dn = fmaxf(cnt[c], 1.0f);
    const float* rowp = csum + (size_t)c * DDIM;

    v8f acc = {0.f, 0.f, 0.f, 0.f, 0.f, 0.f, 0.f, 0.f};
    for (int d0 = 0; d0 < DDIM; d0 += 4) {
        v2f a;
        a[0] = rowp[d0 + klo]     / dn;         // mean[c][d0+K]
        a[1] = rowp[d0 + klo + 1] / dn;
        acc = __builtin_amdgcn_wmma_f32_16x16x4_f32(
            false, a, false, a, (short)0, acc, false, false);
    }
    #pragma unroll
    for (int v = 0; v < 8; ++v) gram[w][v][lane] = acc[v];
    __syncthreads();

    // C layout: VGPR v, lanes 0-15 -> (M=v, N=lane); lanes 16-31 -> (M=v+8, N=lane-16)
    if (lane < 16) {
        const float d = (lane < 8) ? gram[w][lane][lane]
                                   : gram[w][lane - 8][lane + 16];
        normsq[w * 16 + lane] = d;
    }
    __syncthreads();

    float contrib = 0.0f;
    if (tid < DDIM) {
        const float c0  = cnt[tid];
        const float dnn = fmaxf(c0, 1.0f);
        const float var = sm[tid] / dnn - normsq[tid];
        contrib = (c0 / (float)Ntot) * var;
    }
    red[tid] = contrib;
    __syncthreads();
    #pragma unroll
    for (int s = 256; s > 0; s >>= 1) {
        if (tid < s) red[tid] += red[tid + s];
        __syncthreads();
    }
    if (tid == 0) out[0] = red[0];
}

// ---------------------------------------------------------------------------
extern "C" void kernel_launch(void* const* d_in, const int* in_sizes, int n_in,
                              void* d_out, int out_size, void* d_ws, size_t ws_size,
                              hipStream_t stream) {
    const float* emb = (const float*)d_in[0];   // [N, 256]
    const float* co  = (const float*)d_in[1];   // [N, 256]
    const int N = in_sizes[0] / DDIM;

    // Workspace layout (floats): labels[N] | psum[NBLK*65536] | pcnt[NBLK*256]
    //                            | psm[NBLK*256] | csum[65536] | cnt[256] | sm[256]
    float* ws = (float*)d_ws;
    const size_t avail = ws_size / 4;
    long rem = (long)avail - (long)N - (long)(ACC_F + 2 * DDIM);
    int NBLK = (int)(rem / (long)(ACC_F + 2 * DDIM));
    if (NBLK > 128) NBLK = 128;
    if (NBLK < 1)   NBLK = 1;

    int*   labels = (int*)ws;
    float* psum = ws + N;
    float* pcnt = psum + (size_t)NBLK * ACC_F;
    float* psm  = pcnt + (size_t)NBLK * DDIM;
    float* csum = psm  + (size_t)NBLK * DDIM;
    float* cnt  = csum + ACC_F;
    float* sm   = cnt  + DDIM;

    // P1: one wave per row
    {
        const int blocks = (N + 7) / 8;       // 8 waves (256 thr) per block
        k_argmax<<<blocks, 256, 0, stream>>>(co, labels, N);
    }
    // P2: async-staged LDS segment accumulation
    const int ntiles = (N + TILE_ROWS - 1) / TILE_ROWS;
    {
        const int tpb = (ntiles + NBLK - 1) / NBLK;
        k_accum<<<NBLK, 1024, LDS_BYTES, stream>>>(emb, labels, psum, pcnt, psm,
                                                   ntiles, tpb, N);
    }
    // P3: reduce partials
    {
        const int total = ACC_F + 2 * DDIM;
        k_reduce<<<(total + 255) / 256, 256, 0, stream>>>(psum, pcnt, psm,
                                                          csum, cnt, sm, NBLK);
    }
    // P4: WMMA Gram finalize
    k_final<<<1, 512, 0, stream>>>(csum, cnt, sm, (float*)d_out, N);
}